// TopKDistillationLoss_17978733101649
// MI455X (gfx1250) — compile-verified
//
#include <hip/hip_runtime.h>

// Problem constants (from reference setup_inputs)
#define Vv      50257
#define Bb      2
#define Ss      2048
#define NROWS   (Bb * (Ss - 1))      // 4094
#define KK      64
#define INV_T   0.25f                // 1/TEMP
#define T2      16.0                 // TEMP^2
#define THREADS 256
#define NWAVES  (THREADS / 32)

typedef float v2f __attribute__((ext_vector_type(2)));
typedef float v8f __attribute__((ext_vector_type(8)));

// Monotonic uint32 key: larger float -> larger key
__device__ __forceinline__ unsigned keyOf(float x) {
  unsigned u = __float_as_uint(x);
  return (u & 0x80000000u) ? ~u : (u | 0x80000000u);
}

// ---- wave (32-lane) reductions -------------------------------------------
__device__ __forceinline__ float wave_max(float v) {
  v = fmaxf(v, __shfl_xor(v, 16));
  v = fmaxf(v, __shfl_xor(v, 8));
  v = fmaxf(v, __shfl_xor(v, 4));
  v = fmaxf(v, __shfl_xor(v, 2));
  v = fmaxf(v, __shfl_xor(v, 1));
  return v;
}

// 32-lane sum: one v_wmma_f32_16x16x4_f32 with A=ones folds lanes L and L+16
// (D[m][n] = sum_k B[k][n]; B row k0 = lanes0-15, other half-wave = another row),
// then 4 shfl_xor steps finish 16 -> 1. EXEC is all-1s at every call site.
__device__ __forceinline__ float wave_sum_wmma(float v) {
  v2f a; a.x = 1.0f; a.y = 1.0f;   // 16x4 ones
  v2f b; b.x = v;    b.y = 0.0f;   // our 32 partials occupy two B rows
  v8f c = {};
  c = __builtin_amdgcn_wmma_f32_16x16x4_f32(false, a, false, b, (short)0, c,
                                            false, false);
  float h = c[0];                  // lane L: v[L&15] + v[(L&15)+16]
  h += __shfl_xor(h, 8);
  h += __shfl_xor(h, 4);
  h += __shfl_xor(h, 2);
  h += __shfl_xor(h, 1);
  return h;
}

// Ballot/leader-aggregated LDS histogram add: one ds atomic per distinct
// digit among the wave's active lanes (normal data -> ~6 hot bins).
__device__ __forceinline__ void wave_hist_add(unsigned* hist, unsigned d) {
  unsigned active = (unsigned)__ballot(1);
  bool done = false;
  while (active) {
    unsigned leader = (unsigned)__ffs(active) - 1u;
    unsigned lead_d = (unsigned)__shfl((int)d, (int)leader);
    unsigned peers  = (unsigned)__ballot(d == lead_d) & active;
    if (!done && d == lead_d) {
      if ((threadIdx.x & 31u) == leader) atomicAdd(&hist[d], (unsigned)__popc(peers));
      done = true;
    }
    active &= ~peers;
  }
}

// ---- kernels --------------------------------------------------------------
__global__ void tkd_init(double* acc) {
  if (threadIdx.x < 2) acc[threadIdx.x] = 0.0;
}

__global__ __launch_bounds__(THREADS)
void tkd_row(const float* __restrict__ sl, const float* __restrict__ tl,
             const int* __restrict__ tgt, double* __restrict__ acc) {
  __shared__ unsigned hist[NWAVES][256];
  __shared__ unsigned histM[256];
  __shared__ float    redf[NWAVES];
  __shared__ float    sh_bmax, sh_bsum, sh_slabel;
  __shared__ unsigned sh_prefix, sh_krem, sh_cnt;
  __shared__ float    topv[KK];
  __shared__ int      topi[KK];
  __shared__ float    sgth[KK];

  const int tid  = threadIdx.x;
  const int wave = tid >> 5;
  const int row  = blockIdx.x;
  const int b    = row / (Ss - 1);
  const int s    = row - b * (Ss - 1);
  const long base = (long)(b * Ss + s) * (long)Vv;
  const float* __restrict__ srow = sl + base;
  const float* __restrict__ trow = tl + base;

  if (tid == 0) {
    sh_prefix = 0u; sh_krem = KK; sh_cnt = 0u;
    sh_slabel = srow[tgt[b * Ss + s + 1]];
  }
  #pragma unroll
  for (int w = 0; w < NWAVES; ++w) hist[w][tid] = 0u;
  __syncthreads();

  // ---- Pass 1 (HBM): student online logsumexp + teacher byte3 histogram ----
  float m = -INFINITY, es = 0.0f;
  for (int i = tid; i < Vv; i += THREADS) {
    float sv = srow[i];
    float tv = trow[i];
    __builtin_prefetch(srow + i + THREADS * 8, 0, 0);
    __builtin_prefetch(trow + i + THREADS * 8, 0, 0);
    if (sv > m) { es = es * __expf(m - sv) + 1.0f; m = sv; }
    else        { es += __expf(sv - m); }
    wave_hist_add(&hist[wave][0], keyOf(tv) >> 24);
  }

  // block max of student logits
  float wm = wave_max(m);
  if ((tid & 31) == 0) redf[wave] = wm;
  __syncthreads();
  if (tid == 0) {
    float bm = redf[0];
    #pragma unroll
    for (int w = 1; w < NWAVES; ++w) bm = fmaxf(bm, redf[w]);
    sh_bmax = bm;
  }
  __syncthreads();
  const float bm = sh_bmax;
  float contrib = es * __expf(m - bm);
  float ws = wave_sum_wmma(contrib);          // <-- v_wmma fold + shuffles
  if ((tid & 31) == 0) redf[wave] = ws;
  __syncthreads();
  if (tid == 0) {
    float bs = 0.0f;
    #pragma unroll
    for (int w = 0; w < NWAVES; ++w) bs += redf[w];
    sh_bsum = bs;
  }
  // merge per-wave histograms (barrier above covers hist writes)
  unsigned hsum = 0u;
  #pragma unroll
  for (int w = 0; w < NWAVES; ++w) hsum += hist[w][tid];
  histM[tid] = hsum;
  __syncthreads();

  // resolve top byte
  if (tid == 0) {
    unsigned krem = sh_krem, cum = 0u; int d = 255;
    for (; d > 0; --d) { unsigned h = histM[d]; if (cum + h >= krem) break; cum += h; }
    sh_prefix |= ((unsigned)d) << 24;
    sh_krem = krem - cum;
  }
  __syncthreads();

  // ---- Passes 2..4 (L2-hot): refine bytes 2,1,0 ----
  for (int shift = 16; shift >= 0; shift -= 8) {
    #pragma unroll
    for (int w = 0; w < NWAVES; ++w) hist[w][tid] = 0u;
    __syncthreads();
    const unsigned pfx = sh_prefix;
    const int hb = shift + 8;
    for (int i = tid; i < Vv; i += THREADS) {
      unsigned key = keyOf(trow[i]);
      if ((key >> hb) == (pfx >> hb))
        wave_hist_add(&hist[wave][0], (key >> shift) & 255u);
    }
    __syncthreads();
    unsigned hs = 0u;
    #pragma unroll
    for (int w = 0; w < NWAVES; ++w) hs += hist[w][tid];
    histM[tid] = hs;
    __syncthreads();
    if (tid == 0) {
      unsigned krem = sh_krem, cum = 0u; int d = 255;
      for (; d > 0; --d) { unsigned h = histM[d]; if (cum + h >= krem) break; cum += h; }
      sh_prefix |= ((unsigned)d) << shift;
      sh_krem = krem - cum;
    }
    __syncthreads();
  }

  // ---- Collect exactly 64: strictly-greater first, then ties ----
  const unsigned T = sh_prefix;
  for (int i = tid; i < Vv; i += THREADS) {
    float tv = trow[i];
    if (keyOf(tv) > T) {
      unsigned p = atomicAdd(&sh_cnt, 1u);
      topv[p] = tv; topi[p] = i;
    }
  }
  __syncthreads();
  for (int i = tid; i < Vv; i += THREADS) {
    float tv = trow[i];
    if (keyOf(tv) == T) {
      unsigned p = atomicAdd(&sh_cnt, 1u);
      if (p < KK) { topv[p] = tv; topi[p] = i; }
    }
  }
  __syncthreads();

  // gather student at teacher's top-k indices (parallel)
  if (tid < KK) sgth[tid] = srow[topi[tid]];
  __syncthreads();

  // ---- per-row losses (k=64, trivial serial epilogue) ----
  if (tid == 0) {
    float ce = (bm + __logf(sh_bsum)) - sh_slabel;

    float tmax = -INFINITY, smax = -INFINITY;
    #pragma unroll 8
    for (int k = 0; k < KK; ++k) {
      tmax = fmaxf(tmax, topv[k]);
      smax = fmaxf(smax, sgth[k]);
    }
    float tsum = 0.0f, ssum = 0.0f;
    #pragma unroll 8
    for (int k = 0; k < KK; ++k) {
      tsum += __expf((topv[k] - tmax) * INV_T);
      ssum += __expf((sgth[k] - smax) * INV_T);
    }
    const float lt = __logf(tsum), ls = __logf(ssum);
    float kd = 0.0f;
    #pragma unroll 8
    for (int k = 0; k < KK; ++k) {
      float tz = (topv[k] - tmax) * INV_T;
      float sz = (sgth[k] - smax) * INV_T;
      float p  = __expf(tz) / tsum;
      kd += p * ((tz - lt) - (sz - ls));
    }
    atomicAdd(&acc[0], (double)ce);   // global_atomic_add_f64
    atomicAdd(&acc[1], (double)kd);
  }
}

__global__ void tkd_fin(const double* __restrict__ acc, float* __restrict__ out) {
  double ce = acc[0] / (double)NROWS;
  double kd = acc[1] / (double)NROWS * T2;
  out[0] = (float)(0.5 * ce + 0.5 * kd);
  out[1] = (float)ce;
  out[2] = (float)kd;
}

extern "C" void kernel_launch(void* const* d_in, const int* in_sizes, int n_in,
                              void* d_out, int out_size, void* d_ws, size_t ws_size,
                              hipStream_t stream) {
  const float* sl  = (const float*)d_in[0];
  const float* tl  = (const float*)d_in[1];
  const int*   tgt = (const int*)d_in[2];
  double* acc = (double*)d_ws;
  float*  out = (float*)d_out;

  tkd_init<<<1, 64, 0, stream>>>(acc);
  tkd_row<<<NROWS, THREADS, 0, stream>>>(sl, tl, tgt, acc);
  tkd_fin<<<1, 1, 0, stream>>>(acc, out);
}